// SlotODEFunc_20280835572143
// MI455X (gfx1250) — compile-verified
//
#include <hip/hip_runtime.h>

// ---------------------------------------------------------------------------
// SlotODEFunc for MI455X (gfx1250).
// Memory-bound op: 268MB of k/v streamed once (~11.5us @ 23.3TB/s floor).
// All GEMMs use V_WMMA_F32_16X16X4_F32 so compute sits far below that roof.
// Doubly-normalized attention is done in a single streaming pass:
//   attended[n,:] = (sum_m softmax_n(l[:,m]) * v[m,:]) / (sum_m softmax_n + eps)
// k tiles are staged into LDS with GLOBAL_LOAD_ASYNC_TO_LDS_B128 (ASYNCcnt),
// avoiding the VGPR round-trip on the streaming path.
// ---------------------------------------------------------------------------

typedef float v2f __attribute__((ext_vector_type(2)));
typedef float v8f __attribute__((ext_vector_type(8)));

#define NSLOT 16
#define DDIM  128
#define HDIM  256
#define NFEAT 4096
#define BATCH 64
#define KP    132   // pitch for 128-wide LDS rows (multiple of 4, bank-safe)
#define AP    68    // pitch for 64-wide att rows
#define HP    260   // pitch for 256-wide LDS rows
#define EPS_LN  1e-5f
#define EPS_ATT 1e-8f

__device__ __forceinline__ v8f wmma4(v2f a, v2f b, v8f c) {
    // D = A(16x4 f32) * B(4x16 f32) + C(16x16 f32)
    return __builtin_amdgcn_wmma_f32_16x16x4_f32(
        false, a, false, b, (short)0, c, false, false);
}

// CDNA5 async global->LDS copy, 16B per lane, tracked on ASYNCcnt.
__device__ __forceinline__ void async_load_b128(const void* gptr, unsigned lds_addr) {
    asm volatile("global_load_async_to_lds_b128 %0, %1, off"
                 :: "v"(lds_addr), "v"(gptr) : "memory");
}
__device__ __forceinline__ void wait_asynccnt0() {
    asm volatile("s_wait_asynccnt 0" ::: "memory");
}

// ===========================================================================
// Kernel 1: per (split, batch): LN(slots) -> q -> streaming attention partials
// grid = (nsplit, B), block = 128 (4 waves)
// ===========================================================================
__global__ __launch_bounds__(128)
void k1_attend(const float* __restrict__ slots,
               const float* __restrict__ kmat,
               const float* __restrict__ vmat,
               const float* __restrict__ Wq,
               const float* __restrict__ lng,
               const float* __restrict__ lnb,
               float* __restrict__ Upart,
               float* __restrict__ Spart,
               int nsplit)
{
    __shared__ float ks  [64 * KP];      // k tile            [m_local][d]
    __shared__ float sn_s[NSLOT * KP];   // layernormed slots [n][d]
    __shared__ float q_s [NSLOT * KP];   // scaled q          [n][d]
    __shared__ float att_s[NSLOT * AP];  // att tile          [n][m_local]
    __shared__ float s_red[4 * NSLOT];
    __shared__ float mu_s[NSLOT], rs_s[NSLOT];

    const int tid   = threadIdx.x;
    const int lane  = tid & 31;
    const int wave  = tid >> 5;
    const int ln16  = lane & 15;
    const int hgrp  = lane >> 4;         // which 16-lane half
    const int cK    = hgrp * 2;          // A/B-layout K offset for upper half
    const int split = blockIdx.x;
    const int b     = blockIdx.y;
    const int chunk = NFEAT / nsplit;
    const int f_base  = split * chunk;
    const int ngroups = chunk / 64;

    // ---- load slots[b], LayerNorm into sn_s -------------------------------
    const float* sb = slots + (size_t)b * NSLOT * DDIM;
    for (int i = tid; i < NSLOT * DDIM; i += 128)
        sn_s[(i >> 7) * KP + (i & 127)] = sb[i];
    __syncthreads();
    if (tid < NSLOT) {
        float mu = 0.f;
        for (int d = 0; d < DDIM; ++d) mu += sn_s[tid * KP + d];
        mu *= (1.f / DDIM);
        float var = 0.f;
        for (int d = 0; d < DDIM; ++d) {
            float x = sn_s[tid * KP + d] - mu; var += x * x;
        }
        var *= (1.f / DDIM);
        mu_s[tid] = mu;
        rs_s[tid] = rsqrtf(var + EPS_LN);
    }
    __syncthreads();
    for (int i = tid; i < NSLOT * DDIM; i += 128) {
        int n = i >> 7, d = i & 127;
        sn_s[n * KP + d] = (sn_s[n * KP + d] - mu_s[n]) * rs_s[n] * lng[d] + lnb[d];
    }
    __syncthreads();

    // ---- q = snorm @ Wq^T, scale folded in (WMMA, 2 e-tiles per wave) -----
    const float qscale = 0.08838834764831845f;  // D^-0.5
    for (int t = 0; t < 2; ++t) {
        int e0 = (wave * 2 + t) * 16;
        v8f c = {};
        #pragma unroll
        for (int kk = 0; kk < DDIM; kk += 4) {
            v2f a  = *(const v2f*)&sn_s[ln16 * KP + kk + cK];
            v2f bq = *(const v2f*)&Wq[(size_t)(e0 + ln16) * DDIM + kk + cK];
            c = wmma4(a, bq, c);
        }
        #pragma unroll
        for (int r = 0; r < 8; ++r)
            q_s[(r + 8 * hgrp) * KP + e0 + ln16] = c[r] * qscale;
    }

    float S_acc = 0.f;
    v8f u0 = {}, u1 = {};                // U accumulators: d-slices 32w, 32w+16
    __syncthreads();

    // ---- stream over feature groups of 64 ---------------------------------
    for (int g = 0; g < ngroups; ++g) {
        const int f0 = f_base + g * 64;
        const float* kg = kmat + (size_t)b * NFEAT * DDIM + (size_t)f0 * DDIM;
        const float* vg = vmat + (size_t)b * NFEAT * DDIM + (size_t)f0 * DDIM;

        // async global->LDS staging of the 64x128 k tile (b128 per lane)
        for (int i = tid; i < 64 * (DDIM / 4); i += 128) {
            int row = i >> 5, c4 = (i & 31) * 4;
            async_load_b128(kg + (size_t)row * DDIM + c4,
                            (unsigned)(uintptr_t)&ks[row * KP + c4]);
        }
        wait_asynccnt0();
        __syncthreads();

        // Stage 1: logits^T tile [feature, slot] per wave; softmax over slots
        {
            v8f c = {};
            const int mrow = wave * 16 + ln16;
            #pragma unroll
            for (int kk = 0; kk < DDIM; kk += 4) {
                v2f a  = *(const v2f*)&ks[mrow * KP + kk + cK];
                v2f bq = *(const v2f*)&q_s[ln16 * KP + kk + cK];
                c = wmma4(a, bq, c);
            }
            // each C row (fixed feature) lives in one 16-lane half of one VGPR:
            // softmax over the 16 slots == 16-lane xor-shuffle reduction.
            float sl = 0.f;
            #pragma unroll
            for (int r = 0; r < 8; ++r) {
                float x = c[r];
                float m = x;
                m = fmaxf(m, __shfl_xor(m, 1, 16));
                m = fmaxf(m, __shfl_xor(m, 2, 16));
                m = fmaxf(m, __shfl_xor(m, 4, 16));
                m = fmaxf(m, __shfl_xor(m, 8, 16));
                float ev = __expf(x - m);
                float s = ev;
                s += __shfl_xor(s, 1, 16);
                s += __shfl_xor(s, 2, 16);
                s += __shfl_xor(s, 4, 16);
                s += __shfl_xor(s, 8, 16);
                float a = ev / s;
                // transpose through LDS: att_s[slot][feature_local]
                att_s[ln16 * AP + wave * 16 + r + 8 * hgrp] = a;
                sl += a;
            }
            sl += __shfl_xor(sl, 16, 32);   // add other feature half
            S_acc += sl;                    // per-lane: slot = ln16
        }
        __syncthreads();

        // prefetch next k tile while doing the U GEMM
        if (g + 1 < ngroups) {
            const float* kn = kg + 64 * DDIM;
            __builtin_prefetch(kn + (size_t)tid * 32, 0, 0);
            __builtin_prefetch(kn + (size_t)(tid + 128) * 32, 0, 0);
        }

        // Stage 2: U[slot, d-slice] += att_tile @ v_tile  (WMMA, B from global)
        {
            const int d0 = wave * 32;
            #pragma unroll
            for (int kk = 0; kk < 64; kk += 4) {
                v2f a = *(const v2f*)&att_s[ln16 * AP + kk + cK];
                const float* vr0 = vg + (size_t)(kk + cK) * DDIM;
                const float* vr1 = vr0 + DDIM;
                v2f bA, bB;
                bA.x = vr0[d0 + ln16];      bA.y = vr1[d0 + ln16];
                bB.x = vr0[d0 + 16 + ln16]; bB.y = vr1[d0 + 16 + ln16];
                u0 = wmma4(a, bA, u0);
                u1 = wmma4(a, bB, u1);
            }
        }
        // barrier at top of next iteration separates att_s/ks rewrite
    }
    __syncthreads();

    // ---- write partials ----------------------------------------------------
    const size_t pbase = (size_t)b * nsplit + split;
    float* Up = Upart + pbase * (NSLOT * DDIM);
    #pragma unroll
    for (int dt = 0; dt < 2; ++dt) {
        v8f u = dt ? u1 : u0;
        int d = wave * 32 + dt * 16 + ln16;
        #pragma unroll
        for (int r = 0; r < 8; ++r) {
            int n = r + 8 * hgrp;
            Up[n * DDIM + d] = u[r];
        }
    }
    if (lane < 16) s_red[wave * NSLOT + ln16] = S_acc;
    __syncthreads();
    if (tid < NSLOT) {
        float s = s_red[tid] + s_red[NSLOT + tid] +
                  s_red[2 * NSLOT + tid] + s_red[3 * NSLOT + tid];
        Spart[pbase * NSLOT + tid] = s;
    }
}

// ===========================================================================
// Kernel 2: per batch: reduce partials -> GRU (WMMA gates) -> LN -> MLP (WMMA)
// grid = B, block = 256 (8 waves)
// ===========================================================================
__global__ __launch_bounds__(256)
void k2_finalize(const float* __restrict__ slots,
                 const float* __restrict__ W_ih,
                 const float* __restrict__ W_hh,
                 const float* __restrict__ b_ih,
                 const float* __restrict__ b_n,
                 const float* __restrict__ lng,
                 const float* __restrict__ lnb,
                 const float* __restrict__ W0,
                 const float* __restrict__ b0,
                 const float* __restrict__ W1,
                 const float* __restrict__ b1,
                 const float* __restrict__ Upart,
                 const float* __restrict__ Spart,
                 float* __restrict__ out,
                 int nsplit)
{
    __shared__ float att_s[NSLOT * KP];   // attended [n][d]
    __shared__ float sl_s [NSLOT * KP];   // raw slots [n][d]
    __shared__ float g_s  [NSLOT * KP];   // gru_out  [n][d]
    __shared__ float gn_s [NSLOT * KP];   // LN(gru_out) [n][d]
    __shared__ float h_s  [NSLOT * HP];   // relu MLP hidden [n][h]
    __shared__ float s_sh[NSLOT], mu_s[NSLOT], rs_s[NSLOT];

    const int tid  = threadIdx.x;
    const int lane = tid & 31;
    const int wave = tid >> 5;
    const int ln16 = lane & 15;
    const int hgrp = lane >> 4;
    const int cK   = hgrp * 2;
    const int b    = blockIdx.x;

    // ---- load slots, reduce attention partials ----------------------------
    const float* sb = slots + (size_t)b * NSLOT * DDIM;
    for (int i = tid; i < NSLOT * DDIM; i += 256)
        sl_s[(i >> 7) * KP + (i & 127)] = sb[i];
    if (tid < NSLOT) {
        float s = 0.f;
        for (int sp = 0; sp < nsplit; ++sp)
            s += Spart[((size_t)b * nsplit + sp) * NSLOT + tid];
        s_sh[tid] = s + EPS_ATT;
    }
    __syncthreads();
    for (int i = tid; i < NSLOT * DDIM; i += 256) {
        int n = i >> 7, d = i & 127;
        float u = 0.f;
        for (int sp = 0; sp < nsplit; ++sp)
            u += Upart[((size_t)b * nsplit + sp) * (NSLOT * DDIM) + i];
        att_s[n * KP + d] = u / s_sh[n];
    }
    __syncthreads();

    // ---- GRU gates: wave w owns gate-columns j = 16w..16w+15 --------------
    const int j0 = wave * 16;
    v8f cir = {}, cii = {}, cin = {}, chr_ = {}, chi = {}, chn = {};
    #pragma unroll
    for (int kk = 0; kk < DDIM; kk += 4) {
        v2f aa = *(const v2f*)&att_s[ln16 * KP + kk + cK];
        v2f ah = *(const v2f*)&sl_s [ln16 * KP + kk + cK];
        v2f br  = *(const v2f*)&W_ih[(size_t)(      j0 + ln16) * DDIM + kk + cK];
        v2f bi  = *(const v2f*)&W_ih[(size_t)(128 + j0 + ln16) * DDIM + kk + cK];
        v2f bn2 = *(const v2f*)&W_ih[(size_t)(256 + j0 + ln16) * DDIM + kk + cK];
        v2f hr  = *(const v2f*)&W_hh[(size_t)(      j0 + ln16) * DDIM + kk + cK];
        v2f hi  = *(const v2f*)&W_hh[(size_t)(128 + j0 + ln16) * DDIM + kk + cK];
        v2f hn  = *(const v2f*)&W_hh[(size_t)(256 + j0 + ln16) * DDIM + kk + cK];
        cir  = wmma4(aa, br,  cir);
        cii  = wmma4(aa, bi,  cii);
        cin  = wmma4(aa, bn2, cin);
        chr_ = wmma4(ah, hr,  chr_);
        chi  = wmma4(ah, hi,  chi);
        chn  = wmma4(ah, hn,  chn);
    }
    {
        const int j = j0 + ln16;
        const float bir = b_ih[j], bii = b_ih[128 + j], bin = b_ih[256 + j];
        const float bnn = b_n[j];
        #pragma unroll
        for (int r = 0; r < 8; ++r) {
            int n = r + 8 * hgrp;
            float reset = 1.f / (1.f + __expf(-(cir[r] + bir + chr_[r])));
            float inp   = 1.f / (1.f + __expf(-(cii[r] + bii + chi[r])));
            float nv    = tanhf(cin[r] + bin + reset * (chn[r] + bnn));
            float sl    = sl_s[n * KP + j];
            g_s[n * KP + j] = nv + inp * (sl - nv);
        }
    }
    __syncthreads();

    // ---- LayerNorm(gru_out) ------------------------------------------------
    if (tid < NSLOT) {
        float mu = 0.f;
        for (int d = 0; d < DDIM; ++d) mu += g_s[tid * KP + d];
        mu *= (1.f / DDIM);
        float var = 0.f;
        for (int d = 0; d < DDIM; ++d) {
            float x = g_s[tid * KP + d] - mu; var += x * x;
        }
        var *= (1.f / DDIM);
        mu_s[tid] = mu;
        rs_s[tid] = rsqrtf(var + EPS_LN);
    }
    __syncthreads();
    for (int i = tid; i < NSLOT * DDIM; i += 256) {
        int n = i >> 7, d = i & 127;
        gn_s[n * KP + d] = (g_s[n * KP + d] - mu_s[n]) * rs_s[n] * lng[d] + lnb[d];
    }
    __syncthreads();

    // ---- MLP layer 0: h = relu(gn @ W0^T + b0), wave w: tiles 16w, 128+16w -
    for (int t = 0; t < 2; ++t) {
        int e0 = t * 128 + wave * 16;
        v8f c = {};
        #pragma unroll
        for (int kk = 0; kk < DDIM; kk += 4) {
            v2f a  = *(const v2f*)&gn_s[ln16 * KP + kk + cK];
            v2f bb = *(const v2f*)&W0[(size_t)(e0 + ln16) * DDIM + kk + cK];
            c = wmma4(a, bb, c);
        }
        float bias = b0[e0 + ln16];
        #pragma unroll
        for (int r = 0; r < 8; ++r)
            h_s[(r + 8 * hgrp) * HP + e0 + ln16] = fmaxf(c[r] + bias, 0.f);
    }
    __syncthreads();

    // ---- MLP layer 1 + residual + (- slots), wave w: d-tile 16w ------------
    {
        const int d0 = wave * 16;
        v8f c = {};
        #pragma unroll
        for (int kk = 0; kk < HDIM; kk += 4) {
            v2f a  = *(const v2f*)&h_s[ln16 * HP + kk + cK];
            v2f bb = *(const v2f*)&W1[(size_t)(d0 + ln16) * HDIM + kk + cK];
            c = wmma4(a, bb, c);
        }
        const int d = d0 + ln16;
        const float bias = b1[d];
        #pragma unroll
        for (int r = 0; r < 8; ++r) {
            int n = r + 8 * hgrp;
            out[(size_t)b * NSLOT * DDIM + n * DDIM + d] =
                c[r] + bias + g_s[n * KP + d] - sl_s[n * KP + d];
        }
    }
}

// ===========================================================================
extern "C" void kernel_launch(void* const* d_in, const int* in_sizes, int n_in,
                              void* d_out, int out_size, void* d_ws, size_t ws_size,
                              hipStream_t stream)
{
    (void)in_sizes; (void)n_in; (void)out_size;
    const float* slots = (const float*)d_in[1];
    const float* kmat  = (const float*)d_in[2];
    const float* vmat  = (const float*)d_in[3];
    const float* Wq    = (const float*)d_in[4];
    const float* lsg   = (const float*)d_in[5];
    const float* lsb   = (const float*)d_in[6];
    const float* W_ih  = (const float*)d_in[7];
    const float* W_hh  = (const float*)d_in[8];
    const float* b_ih  = (const float*)d_in[9];
    const float* b_n   = (const float*)d_in[10];
    const float* lfg   = (const float*)d_in[11];
    const float* lfb   = (const float*)d_in[12];
    const float* W0    = (const float*)d_in[13];
    const float* b0    = (const float*)d_in[14];
    const float* W1    = (const float*)d_in[15];
    const float* b1    = (const float*)d_in[16];
    float* out = (float*)d_out;

    // pick largest power-of-two feature split that fits in scratch (determ.)
    const size_t per_split = (size_t)BATCH * (NSLOT * DDIM + NSLOT) * sizeof(float);
    int nsplit = 8;
    while (nsplit > 1 && per_split * (size_t)nsplit > ws_size) nsplit >>= 1;

    float* Upart = (float*)d_ws;                                  // [B*ns][16*128]
    float* Spart = Upart + (size_t)BATCH * nsplit * NSLOT * DDIM; // [B*ns][16]

    dim3 g1(nsplit, BATCH);
    k1_attend<<<g1, 128, 0, stream>>>(slots, kmat, vmat, Wq, lsg, lsb,
                                      Upart, Spart, nsplit);
    k2_finalize<<<BATCH, 256, 0, stream>>>(slots, W_ih, W_hh, b_ih, b_n,
                                           lfg, lfb, W0, b0, W1, b1,
                                           Upart, Spart, out, nsplit);
}